// GATConcDQL_695784702655
// MI455X (gfx1250) — compile-verified
//
#include <hip/hip_runtime.h>
#include <hip/hip_bf16.h>

// ---------------- problem constants ----------------
#define NN    131072      // nodes
#define EE    2097152     // edges
#define BB    64          // graphs
#define NPGc  2048        // nodes per graph
#define KKc   1024        // kept per graph
#define NN2   65536       // pooled nodes
#define LEAKC 0.2f
#define TPB   256

typedef __attribute__((ext_vector_type(2))) float v2f;
typedef __attribute__((ext_vector_type(8))) float v8f;

// order-preserving float<->uint mapping for atomicMax-based segment max
__device__ __forceinline__ unsigned fOrderKey(float f) {
  unsigned u = __float_as_uint(f);
  return (u & 0x80000000u) ? ~u : (u | 0x80000000u);
}
__device__ __forceinline__ float fOrderInv(unsigned u) {
  unsigned b = (u & 0x80000000u) ? (u & 0x7FFFFFFFu) : ~u;
  return __uint_as_float(b);
}

// ---------------- WMMA dual linear: gl = X@Wl^T+bl, gr = X@Wr^T+br ----------
// X:[Nn,CIN] f32, W:[5,CIN], out:[Nn,5]. One wave per 16-row tile,
// V_WMMA_F32_16X16X4_F32 accumulating over K chunks of 4 (exact fp32).
// CIN is a compile-time constant so the chunk loop fully unrolls with
// branchless (clamp+mask) edge handling -> no divergent EXEC-masked loads.
template <int CIN>
__global__ void k_lin2_wmma(const float* __restrict__ X,
                            const float* __restrict__ Wl, const float* __restrict__ bl,
                            const float* __restrict__ Wr, const float* __restrict__ br,
                            float* __restrict__ gl, float* __restrict__ gr) {
  int wave = (blockIdx.x * blockDim.x + threadIdx.x) >> 5;
  int lane = threadIdx.x & 31;
  int half = lane >> 4;       // 0: K pair (k0,k0+1), 1: K pair (k0+2,k0+3)
  int l15  = lane & 15;
  int row0 = wave * 16;
  constexpr int NCH = (CIN + 3) / 4;
  int rowA = row0 + l15;
  int wrow = (l15 < 5) ? l15 : 0;      // clamp B row, mask below
  float nm = (l15 < 5) ? 1.0f : 0.0f;
  v8f cl = {0.f,0.f,0.f,0.f,0.f,0.f,0.f,0.f};
  v8f cr = cl;
#pragma unroll
  for (int c = 0; c < NCH; ++c) {
    int k0 = c * 4 + half * 2;
    v2f a, b1, b2;
    if constexpr ((CIN % 4) == 0) {
      // k0, k0+1 always in range; 8-byte aligned vector loads
      a  = *(const v2f*)(X  + rowA * CIN + k0);
      b1 = *(const v2f*)(Wl + wrow * CIN + k0) * nm;
      b2 = *(const v2f*)(Wr + wrow * CIN + k0) * nm;
    } else {
      int ka = (k0     < CIN) ? k0     : 0;
      int kb = (k0 + 1 < CIN) ? k0 + 1 : 0;
      float m0 = (k0     < CIN) ? 1.0f : 0.0f;
      float m1 = (k0 + 1 < CIN) ? 1.0f : 0.0f;
      a  = (v2f){X[rowA * CIN + ka] * m0,       X[rowA * CIN + kb] * m1};
      b1 = (v2f){Wl[wrow * CIN + ka] * (m0*nm), Wl[wrow * CIN + kb] * (m1*nm)};
      b2 = (v2f){Wr[wrow * CIN + ka] * (m0*nm), Wr[wrow * CIN + kb] * (m1*nm)};
    }
    cl = __builtin_amdgcn_wmma_f32_16x16x4_f32(false, a, false, b1, (short)0, cl, false, false);
    cr = __builtin_amdgcn_wmma_f32_16x16x4_f32(false, a, false, b2, (short)0, cr, false, false);
  }
  // D 16x16: VGPR v -> row0+v (lanes0-15) / row0+8+v (lanes16-31), col = l15
  int col = l15;
  if (col < 5) {
    float bL = bl[col], bR = br[col];
#pragma unroll
    for (int v = 0; v < 8; ++v) {
      int row = row0 + v + half * 8;
      gl[row * 5 + col] = cl[v] + bL;
      gr[row * 5 + col] = cr[v] + bR;
    }
  }
}

// --------- self-loop attr accumulation: loopA[d] += eattr, deg[d] += 1 -------
__global__ void k_loop_accum(const int* __restrict__ src, const int* __restrict__ dst,
                             const float* __restrict__ eattr, const int* __restrict__ remap,
                             float* __restrict__ loopA, float* __restrict__ deg) {
  int e = blockIdx.x * blockDim.x + threadIdx.x;
  if (e >= EE) return;
  __builtin_prefetch(eattr + (size_t)(e < EE - 4096 ? e + 4096 : e) * 8, 0, 0);
  int d = dst[e];
  if (remap) {
    int s = remap[src[e]];
    d = remap[d];
    if (s < 0 || d < 0) return;   // masked edge (not between kept nodes)
  }
  const float4* ea = (const float4*)(eattr + (size_t)e * 8);
  float4 e0 = ea[0], e1 = ea[1];
  float ev[8] = {e0.x, e0.y, e0.z, e0.w, e1.x, e1.y, e1.z, e1.w};
#pragma unroll
  for (int k = 0; k < 8; ++k) atomicAdd(&loopA[d * 8 + k], ev[k]);
  atomicAdd(&deg[d], 1.0f);
}

__global__ void k_loop_div(float* __restrict__ loopA, const float* __restrict__ deg, int n) {
  int i = blockIdx.x * blockDim.x + threadIdx.x;
  if (i >= n) return;
  float c = 1.0f / fmaxf(deg[i], 1.0f);
  float4* row = (float4*)(loopA + (size_t)i * 8);
  float4 a = row[0], b = row[1];
  a.x *= c; a.y *= c; a.z *= c; a.w *= c;
  b.x *= c; b.y *= c; b.z *= c; b.w *= c;
  row[0] = a; row[1] = b;
}

// --------- per-edge attention logit (E real edges + Nloop self loops) --------
__global__ void k_edge_logit(const int* __restrict__ src, const int* __restrict__ dst,
                             const float* __restrict__ eattr, const float* __restrict__ loopA,
                             const float* __restrict__ gl, const float* __restrict__ gr,
                             const float* __restrict__ We, const float* __restrict__ att,
                             const int* __restrict__ remap, int Nloop,
                             float* __restrict__ logit) {
  int tid = blockIdx.x * blockDim.x + threadIdx.x;
  if (tid >= EE + Nloop) return;
  int s, d;
  const float* ea;
  if (tid < EE) {
    s = src[tid]; d = dst[tid];
    if (remap) {
      s = remap[s]; d = remap[d];
      if (s < 0 || d < 0) { logit[tid] = -1e30f; return; }
    }
    ea = eattr + (size_t)tid * 8;
    __builtin_prefetch(eattr + (size_t)(tid < EE - 4096 ? tid + 4096 : tid) * 8, 0, 0);
  } else {
    int n = tid - EE;
    s = n; d = n;
    ea = loopA + (size_t)n * 8;
  }
  float4 e0 = ((const float4*)ea)[0];
  float4 e1 = ((const float4*)ea)[1];
  float ev[8] = {e0.x, e0.y, e0.z, e0.w, e1.x, e1.y, e1.z, e1.w};
  float acc = 0.f;
#pragma unroll
  for (int j = 0; j < 5; ++j) {
    float v = 0.f;
#pragma unroll
    for (int k = 0; k < 8; ++k) v += ev[k] * We[j * 8 + k];
    float z = gl[s * 5 + j] + gr[d * 5 + j] + v;
    z = (z >= 0.f) ? z : LEAKC * z;
    acc += z * att[j];
  }
  logit[tid] = acc;
}

__global__ void k_seg_max(const float* __restrict__ logit, const int* __restrict__ dst,
                          const int* __restrict__ remap, int Nloop, unsigned* __restrict__ mx) {
  int tid = blockIdx.x * blockDim.x + threadIdx.x;
  if (tid >= EE + Nloop) return;
  int d;
  if (tid < EE) {
    d = dst[tid];
    if (remap) { d = remap[d]; if (d < 0) return; }
  } else d = tid - EE;
  atomicMax(&mx[d], fOrderKey(logit[tid]));
}

__global__ void k_exp_sum(const float* __restrict__ logit, const int* __restrict__ dst,
                          const int* __restrict__ remap, int Nloop,
                          const unsigned* __restrict__ mx,
                          float* __restrict__ wexp, float* __restrict__ denom) {
  int tid = blockIdx.x * blockDim.x + threadIdx.x;
  if (tid >= EE + Nloop) return;
  int d;
  if (tid < EE) {
    d = dst[tid];
    if (remap) { d = remap[d]; if (d < 0) { wexp[tid] = 0.f; return; } }
  } else d = tid - EE;
  float m = fOrderInv(mx[d]);
  float w = expf(logit[tid] - m);
  wexp[tid] = w;
  atomicAdd(&denom[d], w);
}

__global__ void k_scatter(const float* __restrict__ wexp, const float* __restrict__ denom,
                          const int* __restrict__ src, const int* __restrict__ dst,
                          const int* __restrict__ remap, int Nloop,
                          const float* __restrict__ gl,
                          float* __restrict__ hout, float* __restrict__ natt) {
  int tid = blockIdx.x * blockDim.x + threadIdx.x;
  if (tid >= EE + Nloop) return;
  int s, d;
  if (tid < EE) {
    s = src[tid]; d = dst[tid];
    if (remap) { s = remap[s]; d = remap[d]; if (s < 0 || d < 0) return; }
  } else { s = tid - EE; d = s; }
  float alpha = wexp[tid] / (denom[d] + 1e-16f);
#pragma unroll
  for (int j = 0; j < 5; ++j) atomicAdd(&hout[d * 5 + j], alpha * gl[s * 5 + j]);
  if (natt) atomicAdd(&natt[s], alpha);
}

// --------- GraphNorm (with conv bias fused in) + ReLU, optional mean pool ---
__global__ void k_graphnorm(const float* __restrict__ h, const float* __restrict__ bias,
                            const float* __restrict__ gw, const float* __restrict__ gb,
                            const float* __restrict__ gms, int npg,
                            float* __restrict__ out, float* __restrict__ gpool) {
  __shared__ float red[5][TPB];
  int b = blockIdx.x, t = threadIdx.x;
  int base = b * npg;
  float bj[5], wj[5], b2j[5], msj[5];
  for (int j = 0; j < 5; ++j) { bj[j] = bias[j]; wj[j] = gw[j]; b2j[j] = gb[j]; msj[j] = gms[j]; }
  float acc[5] = {0.f,0.f,0.f,0.f,0.f};
  for (int i = t; i < npg; i += TPB)
    for (int j = 0; j < 5; ++j) acc[j] += h[(base + i) * 5 + j] + bj[j];
  for (int j = 0; j < 5; ++j) red[j][t] = acc[j];
  __syncthreads();
  for (int s = TPB / 2; s > 0; s >>= 1) {
    if (t < s) for (int j = 0; j < 5; ++j) red[j][t] += red[j][t + s];
    __syncthreads();
  }
  float mean[5];
  for (int j = 0; j < 5; ++j) mean[j] = red[j][0] / (float)npg;
  __syncthreads();
  for (int j = 0; j < 5; ++j) acc[j] = 0.f;
  for (int i = t; i < npg; i += TPB)
    for (int j = 0; j < 5; ++j) {
      float d = h[(base + i) * 5 + j] + bj[j] - msj[j] * mean[j];
      acc[j] += d * d;
    }
  for (int j = 0; j < 5; ++j) red[j][t] = acc[j];
  __syncthreads();
  for (int s = TPB / 2; s > 0; s >>= 1) {
    if (t < s) for (int j = 0; j < 5; ++j) red[j][t] += red[j][t + s];
    __syncthreads();
  }
  float rstd[5];
  for (int j = 0; j < 5; ++j) rstd[j] = rsqrtf(red[j][0] / (float)npg + 1e-5f);
  __syncthreads();
  for (int j = 0; j < 5; ++j) acc[j] = 0.f;
  for (int i = t; i < npg; i += TPB)
    for (int j = 0; j < 5; ++j) {
      float d = h[(base + i) * 5 + j] + bj[j] - msj[j] * mean[j];
      float y = fmaxf(wj[j] * d * rstd[j] + b2j[j], 0.f);
      out[(base + i) * 5 + j] = y;
      acc[j] += y;
    }
  if (gpool) {
    for (int j = 0; j < 5; ++j) red[j][t] = acc[j];
    __syncthreads();
    for (int s = TPB / 2; s > 0; s >>= 1) {
      if (t < s) for (int j = 0; j < 5; ++j) red[j][t] += red[j][t + s];
      __syncthreads();
    }
    if (t == 0) for (int j = 0; j < 5; ++j) gpool[b * 5 + j] = red[j][0] / (float)npg;
  }
}

// --------- TopK pooling score ------------------------------------------------
__global__ void k_score(const float* __restrict__ natt, const float* __restrict__ pw,
                        float* __restrict__ score) {
  int i = blockIdx.x * blockDim.x + threadIdx.x;
  if (i >= NN) return;
  float s = 0.f, q = 0.f;
#pragma unroll
  for (int j = 0; j < 5; ++j) { float w = pw[j]; s += w; q += w * w; }
  float scale = s / (sqrtf(q) + 1e-16f);
  score[i] = tanhf(natt[i] * scale);
}

// --------- TopK selection per graph: bitonic sort in LDS + scan compaction --
__global__ void k_topk(const float* __restrict__ score, const float* __restrict__ h2n,
                       float* __restrict__ xp, int* __restrict__ gi, int* __restrict__ nidx) {
  __shared__ unsigned long long keys[NPGc];
  __shared__ int keptf[NPGc];
  __shared__ int psA[NPGc];
  __shared__ int psB[NPGc];
  int b = blockIdx.x, t = threadIdx.x;
  int base = b * NPGc;
  for (int i = t; i < NPGc; i += TPB) {
    unsigned u = fOrderKey(score[base + i]);
    // secondary key: lower original index wins ties (matches jax.lax.top_k)
    keys[i] = ((unsigned long long)u << 32) | (unsigned)(NPGc - 1 - i);
  }
  __syncthreads();
  for (int k2 = 2; k2 <= NPGc; k2 <<= 1) {
    for (int j = k2 >> 1; j > 0; j >>= 1) {
      for (int i = t; i < NPGc; i += TPB) {
        int ixj = i ^ j;
        if (ixj > i) {
          bool up = ((i & k2) == 0);   // ascending overall
          unsigned long long a = keys[i], c = keys[ixj];
          if ((a > c) == up) { keys[i] = c; keys[ixj] = a; }
        }
      }
      __syncthreads();
    }
  }
  // top K = last K entries of ascending sort
  for (int i = t; i < NPGc; i += TPB) {
    int li = NPGc - 1 - (int)(unsigned)(keys[i] & 0xFFFFFFFFull);
    keptf[li] = (i >= NPGc - KKc) ? 1 : 0;
  }
  __syncthreads();
  int* cur = psA; int* nxt = psB;
  for (int i = t; i < NPGc; i += TPB) cur[i] = keptf[i];
  __syncthreads();
  for (int off = 1; off < NPGc; off <<= 1) {
    for (int i = t; i < NPGc; i += TPB) nxt[i] = cur[i] + ((i >= off) ? cur[i - off] : 0);
    __syncthreads();
    int* tmp = cur; cur = nxt; nxt = tmp;
  }
  for (int i = t; i < NPGc; i += TPB) {
    int g = base + i;
    if (keptf[i]) {
      int gp = b * KKc + (cur[i] - 1);
      gi[gp] = g;
      nidx[g] = gp;
      float sc = score[g];
      for (int j = 0; j < 5; ++j) xp[gp * 5 + j] = h2n[g * 5 + j] * sc;
    } else {
      nidx[g] = -1;
    }
  }
}

// --------- dueling heads ----------------------------------------------------
__global__ void k_heads(const float* __restrict__ x, const float* __restrict__ x1,
                        const float* __restrict__ gpool, const float* __restrict__ goal,
                        const int* __restrict__ cur, const int* __restrict__ amask,
                        const float* __restrict__ embW, const float* __restrict__ embB,
                        const float* __restrict__ asW, const float* __restrict__ asB,
                        const float* __restrict__ aW,  const float* __restrict__ aB,
                        const float* __restrict__ vsW, const float* __restrict__ vsB,
                        const float* __restrict__ vW,  const float* __restrict__ vB,
                        float* __restrict__ q) {
  int b = threadIdx.x;
  if (b >= BB) return;
  int gidx = cur[b] + KKc * b;   // post-pooling offset as in reference
  float feat[25];
  for (int j = 0; j < 5; ++j) {
    float acc = embB[j];
    for (int k = 0; k < 16; ++k) acc += x[gidx * 16 + k] * embW[j * 16 + k];
    feat[j] = fmaxf(acc, 0.f);
  }
  for (int j = 0; j < 5; ++j)  feat[5 + j]  = x1[gidx * 5 + j];
  for (int j = 0; j < 5; ++j)  feat[10 + j] = gpool[b * 5 + j];
  for (int j = 0; j < 10; ++j) feat[15 + j] = goal[b * 10 + j];
  float hA[16];
  for (int r = 0; r < 16; ++r) {
    float acc = asB[r];
    for (int c = 0; c < 25; ++c) acc += feat[c] * asW[r * 25 + c];
    hA[r] = fmaxf(acc, 0.f);
  }
  float adv[32]; float am = 0.f;
  for (int r = 0; r < 32; ++r) {
    float acc = aB[r];
    for (int c = 0; c < 16; ++c) acc += hA[c] * aW[r * 16 + c];
    adv[r] = acc; am += acc;
  }
  am *= (1.f / 32.f);
  float hV[20];
  for (int r = 0; r < 20; ++r) {
    float acc = vsB[r];
    for (int c = 0; c < 25; ++c) acc += feat[c] * vsW[r * 25 + c];
    hV[r] = fmaxf(acc, 0.f);
  }
  float val = vB[0];
  for (int c = 0; c < 20; ++c) val += hV[c] * vW[c];
  for (int r = 0; r < 32; ++r) {
    float qq = val + adv[r] - am;
    q[b * 32 + r] = (amask[b * 32 + r] == 0) ? -1e8f : qq;
  }
}

// ============================================================================
extern "C" void kernel_launch(void* const* d_in, const int* in_sizes, int n_in,
                              void* d_out, int out_size, void* d_ws, size_t ws_size,
                              hipStream_t stream) {
  (void)in_sizes; (void)n_in; (void)out_size; (void)ws_size;
  const float* x      = (const float*)d_in[0];
  const int*   esrc   = (const int*)  d_in[1];
  const int*   edst   = (const int*)  d_in[2];
  const float* eattr  = (const float*)d_in[3];
  /* d_in[4] = batch (implicit: contiguous graphs) */
  const int*   curids = (const int*)  d_in[5];
  const int*   amask  = (const int*)  d_in[6];
  const float* goal   = (const float*)d_in[7];
  int pi = 8;
  const float* embW = (const float*)d_in[pi++]; const float* embB = (const float*)d_in[pi++];
  const float* c1Wl = (const float*)d_in[pi++]; const float* c1bl = (const float*)d_in[pi++];
  const float* c1Wr = (const float*)d_in[pi++]; const float* c1br = (const float*)d_in[pi++];
  const float* c1We = (const float*)d_in[pi++]; const float* c1att= (const float*)d_in[pi++];
  const float* c1bi = (const float*)d_in[pi++];
  const float* n1w  = (const float*)d_in[pi++]; const float* n1b  = (const float*)d_in[pi++];
  const float* n1ms = (const float*)d_in[pi++];
  const float* c2Wl = (const float*)d_in[pi++]; const float* c2bl = (const float*)d_in[pi++];
  const float* c2Wr = (const float*)d_in[pi++]; const float* c2br = (const float*)d_in[pi++];
  const float* c2We = (const float*)d_in[pi++]; const float* c2att= (const float*)d_in[pi++];
  const float* c2bi = (const float*)d_in[pi++];
  const float* n2w  = (const float*)d_in[pi++]; const float* n2b  = (const float*)d_in[pi++];
  const float* n2ms = (const float*)d_in[pi++];
  const float* poolw= (const float*)d_in[pi++];
  const float* c3Wl = (const float*)d_in[pi++]; const float* c3bl = (const float*)d_in[pi++];
  const float* c3Wr = (const float*)d_in[pi++]; const float* c3br = (const float*)d_in[pi++];
  const float* c3We = (const float*)d_in[pi++]; const float* c3att= (const float*)d_in[pi++];
  const float* c3bi = (const float*)d_in[pi++];
  const float* n3w  = (const float*)d_in[pi++]; const float* n3b  = (const float*)d_in[pi++];
  const float* n3ms = (const float*)d_in[pi++];
  const float* vsW  = (const float*)d_in[pi++]; const float* vsB  = (const float*)d_in[pi++];
  const float* vW   = (const float*)d_in[pi++]; const float* vB   = (const float*)d_in[pi++];
  const float* asW  = (const float*)d_in[pi++]; const float* asB  = (const float*)d_in[pi++];
  const float* aW   = (const float*)d_in[pi++]; const float* aB   = (const float*)d_in[pi++];

  // -------- workspace layout --------
  float* ws = (float*)d_ws;
  size_t off = 0;
  auto alloc = [&](size_t n) { float* r = ws + off; off += n; return r; };
  float*    gl    = alloc((size_t)NN * 5);
  float*    gr    = alloc((size_t)NN * 5);
  float*    loopA = alloc((size_t)NN * 8);
  float*    deg   = alloc(NN);
  float*    logit = alloc((size_t)EE + NN);
  float*    wexp  = alloc((size_t)EE + NN);
  unsigned* mx    = (unsigned*)alloc(NN);
  float*    denom = alloc(NN);
  float*    hbuf  = alloc((size_t)NN * 5);
  float*    x1    = alloc((size_t)NN * 5);
  float*    h2n   = alloc((size_t)NN * 5);
  float*    h3n   = alloc((size_t)NN2 * 5);
  float*    natt  = alloc(NN);
  float*    score = alloc(NN);
  float*    xp    = alloc((size_t)NN2 * 5);
  float*    gpool = alloc((size_t)BB * 5);
  int*      gi    = (int*)alloc(NN2);
  int*      nidx  = (int*)alloc(NN);

  dim3 blk(TPB);
  dim3 gE((EE + TPB - 1) / TPB);
  dim3 gEN1((EE + NN + TPB - 1) / TPB);
  dim3 gEN3((EE + NN2 + TPB - 1) / TPB);
  dim3 gN((NN + TPB - 1) / TPB);
  dim3 gN2((NN2 + TPB - 1) / TPB);
  dim3 gLin1(NN / 128);   // 16 rows per wave, 8 waves per block
  dim3 gLin3(NN2 / 128);

  // ================= conv1 =================
  hipMemsetAsync(loopA, 0, (size_t)NN * 8 * sizeof(float), stream);
  hipMemsetAsync(deg,   0, (size_t)NN * sizeof(float), stream);
  hipMemsetAsync(mx,    0, (size_t)NN * sizeof(unsigned), stream);
  hipMemsetAsync(denom, 0, (size_t)NN * sizeof(float), stream);
  hipMemsetAsync(hbuf,  0, (size_t)NN * 5 * sizeof(float), stream);
  k_loop_accum<<<gE, blk, 0, stream>>>(esrc, edst, eattr, nullptr, loopA, deg);
  k_loop_div<<<gN, blk, 0, stream>>>(loopA, deg, NN);
  k_lin2_wmma<16><<<gLin1, blk, 0, stream>>>(x, c1Wl, c1bl, c1Wr, c1br, gl, gr);
  k_edge_logit<<<gEN1, blk, 0, stream>>>(esrc, edst, eattr, loopA, gl, gr, c1We, c1att, nullptr, NN, logit);
  k_seg_max<<<gEN1, blk, 0, stream>>>(logit, edst, nullptr, NN, mx);
  k_exp_sum<<<gEN1, blk, 0, stream>>>(logit, edst, nullptr, NN, mx, wexp, denom);
  k_scatter<<<gEN1, blk, 0, stream>>>(wexp, denom, esrc, edst, nullptr, NN, gl, hbuf, nullptr);
  k_graphnorm<<<BB, blk, 0, stream>>>(hbuf, c1bi, n1w, n1b, n1ms, NPGc, x1, nullptr);

  // ================= conv2 =================
  hipMemsetAsync(mx,    0, (size_t)NN * sizeof(unsigned), stream);
  hipMemsetAsync(denom, 0, (size_t)NN * sizeof(float), stream);
  hipMemsetAsync(hbuf,  0, (size_t)NN * 5 * sizeof(float), stream);
  hipMemsetAsync(natt,  0, (size_t)NN * sizeof(float), stream);
  k_lin2_wmma<5><<<gLin1, blk, 0, stream>>>(x1, c2Wl, c2bl, c2Wr, c2br, gl, gr);
  k_edge_logit<<<gEN1, blk, 0, stream>>>(esrc, edst, eattr, loopA, gl, gr, c2We, c2att, nullptr, NN, logit);
  k_seg_max<<<gEN1, blk, 0, stream>>>(logit, edst, nullptr, NN, mx);
  k_exp_sum<<<gEN1, blk, 0, stream>>>(logit, edst, nullptr, NN, mx, wexp, denom);
  k_scatter<<<gEN1, blk, 0, stream>>>(wexp, denom, esrc, edst, nullptr, NN, gl, hbuf, natt);
  k_graphnorm<<<BB, blk, 0, stream>>>(hbuf, c2bi, n2w, n2b, n2ms, NPGc, h2n, nullptr);

  // ================= TopK pooling =================
  k_score<<<gN, blk, 0, stream>>>(natt, poolw, score);
  k_topk<<<BB, blk, 0, stream>>>(score, h2n, xp, gi, nidx);

  // ================= conv3 (pooled graph) =================
  hipMemsetAsync(loopA, 0, (size_t)NN2 * 8 * sizeof(float), stream);
  hipMemsetAsync(deg,   0, (size_t)NN2 * sizeof(float), stream);
  hipMemsetAsync(mx,    0, (size_t)NN2 * sizeof(unsigned), stream);
  hipMemsetAsync(denom, 0, (size_t)NN2 * sizeof(float), stream);
  hipMemsetAsync(hbuf,  0, (size_t)NN2 * 5 * sizeof(float), stream);
  k_loop_accum<<<gE, blk, 0, stream>>>(esrc, edst, eattr, nidx, loopA, deg);
  k_loop_div<<<gN2, blk, 0, stream>>>(loopA, deg, NN2);
  k_lin2_wmma<5><<<gLin3, blk, 0, stream>>>(xp, c3Wl, c3bl, c3Wr, c3br, gl, gr);
  k_edge_logit<<<gEN3, blk, 0, stream>>>(esrc, edst, eattr, loopA, gl, gr, c3We, c3att, nidx, NN2, logit);
  k_seg_max<<<gEN3, blk, 0, stream>>>(logit, edst, nidx, NN2, mx);
  k_exp_sum<<<gEN3, blk, 0, stream>>>(logit, edst, nidx, NN2, mx, wexp, denom);
  k_scatter<<<gEN3, blk, 0, stream>>>(wexp, denom, esrc, edst, nidx, NN2, gl, hbuf, nullptr);
  k_graphnorm<<<BB, blk, 0, stream>>>(hbuf, c3bi, n3w, n3b, n3ms, KKc, h3n, gpool);

  // ================= heads =================
  k_heads<<<1, 64, 0, stream>>>(x, x1, gpool, goal, curids, amask,
                                embW, embB, asW, asB, aW, aB, vsW, vsB, vW, vB,
                                (float*)d_out);
}